// MyModel_61933428412916
// MI455X (gfx1250) — compile-verified
//
#include <hip/hip_runtime.h>

// Scatter-add segment sum for MI455X (gfx1250).
// out[0..N)  = segment_sum(values, cols)   (atomic fadd into L2-resident bins)
// out[N..2N) = copy of out[0..N)           (reference returns the tuple twice)

typedef __attribute__((ext_vector_type(4))) float v4f;
typedef __attribute__((ext_vector_type(4))) int   v4i;

__global__ void seg_zero_kernel(float* __restrict__ out, int n) {
    int i = blockIdx.x * blockDim.x + threadIdx.x;
    if (i < n) out[i] = 0.0f;
}

__global__ void seg_scatter_add_kernel(const float* __restrict__ vals,
                                       const int*   __restrict__ cols,
                                       float*       __restrict__ out,
                                       int nnz) {
    const int stride = gridDim.x * blockDim.x;
    const int tid    = blockIdx.x * blockDim.x + threadIdx.x;
    const int nvec   = nnz >> 2;              // 128-bit packets

    const v4f* __restrict__ vv = (const v4f*)vals;
    const v4i* __restrict__ cv = (const v4i*)cols;

    for (int i = tid; i < nvec; i += stride) {
        // Streaming (read-once) data: non-temporal b128 loads.
        v4f v = __builtin_nontemporal_load(&vv[i]);
        v4i c = __builtin_nontemporal_load(&cv[i]);
        // Native global_atomic_add_f32 (no CAS loop); bins live in L2.
        unsafeAtomicAdd(&out[c.x], v.x);
        unsafeAtomicAdd(&out[c.y], v.y);
        unsafeAtomicAdd(&out[c.z], v.z);
        unsafeAtomicAdd(&out[c.w], v.w);
    }

    // Scalar tail (nnz not a multiple of 4).
    for (int i = (nvec << 2) + tid; i < nnz; i += stride) {
        unsafeAtomicAdd(&out[cols[i]], vals[i]);
    }
}

__global__ void seg_dup_kernel(float* __restrict__ out, int n) {
    int i = blockIdx.x * blockDim.x + threadIdx.x;
    if (i < n) out[n + i] = out[i];
}

extern "C" void kernel_launch(void* const* d_in, const int* in_sizes, int n_in,
                              void* d_out, int out_size, void* d_ws, size_t ws_size,
                              hipStream_t stream) {
    const float* vals = (const float*)d_in[0];
    // d_in[1] = rows: unused (reference reduces over dim 0 -> keyed by cols).
    const int*   cols = (const int*)d_in[2];
    // d_in[3] = num_segments scalar lives on device; recover N from out_size.
    const int nnz = in_sizes[0];
    const int n   = out_size / 2;
    float* out = (float*)d_out;

    constexpr int T = 256;

    // 1) Zero the full 2N output (harness poisons it with 0xAA).
    seg_zero_kernel<<<(out_size + T - 1) / T, T, 0, stream>>>(out, out_size);

    // 2) Scatter-add. 2048 blocks x 256 threads (wave32: 8 waves/block) ->
    //    512K threads, ~10 vec4 grid-stride iterations each over 5M packets.
    seg_scatter_add_kernel<<<2048, T, 0, stream>>>(vals, cols, out, nnz);

    // 3) Duplicate first half into second half (tuple's second output).
    seg_dup_kernel<<<(n + T - 1) / T, T, 0, stream>>>(out, n);
}